// Model_83932250898658
// MI455X (gfx1250) — compile-verified
//
#include <hip/hip_runtime.h>
#include <hip/hip_bf16.h>

// ---------------------------------------------------------------------------
// PointNet++ (MSG) forward for MI455X / gfx1250.
// All matmuls run on V_WMMA_F32_16X16X4_F32 (fp32 in/out, matching reference
// numerics). Every GEMM call site in this network has M % 16 == 0 and
// N % 16 == 0, so only the K dimension (cin = 6/323/515/...) needs a masked
// tail step. Each wave computes a 16(M) x 64(N) output tile (4 accumulators),
// reusing its A fragment across 4 WMMAs per k-step. Edge N-tiles are handled
// by clamping the tile index (redundant-but-harmless WMMAs) so the inner loop
// is branch-free and EXEC stays all-ones across every WMMA (ISA 7.12 req).
// Fragment layouts per CDNA5 ISA 7.12.2:
//   A (16x4 f32): lane L holds row M=L&15; VGPR v holds K = 2*(L>>4)+v
//   B ( 4x16 f32): lane L holds col N=L&15; VGPR v holds K = 2*(L>>4)+v
//   C/D (16x16 f32): lane L holds col N=L&15; VGPR v holds row M = v+8*(L>>4)
// ---------------------------------------------------------------------------

typedef float v2f __attribute__((ext_vector_type(2)));
typedef float v8f __attribute__((ext_vector_type(8)));

static __device__ __forceinline__ v8f wmma4(v2f a, v2f b, v8f c) {
  return __builtin_amdgcn_wmma_f32_16x16x4_f32(false, a, false, b, (short)0, c,
                                               false, false);
}

static __device__ __forceinline__ void store_tile(float* __restrict__ Y,
                                                  const v8f& acc,
                                                  const float* __restrict__ sc,
                                                  const float* __restrict__ bi,
                                                  int N, int act, int mBase,
                                                  int nCol) {
  float s = sc ? sc[nCol] : 1.f;
  float bb = bi ? bi[nCol] : 0.f;
#pragma unroll
  for (int v = 0; v < 8; ++v) {
    float y = acc[v] * s + bb;
    if (act == 1) y = fmaxf(y, 0.f);
    Y[(size_t)(mBase + v) * N + nCol] = y;
  }
}

// ------------------------------ GEMM + affine + act ------------------------
// Y[m,n] = act( (A[m,:] @ W[:,n]) * scale[n] + bias[n] )
// Preconditions: M % 16 == 0, N % 16 == 0. act: 0 = identity, 1 = relu.
__global__ __launch_bounds__(128) void gemm_affine_act_kernel(
    const float* __restrict__ A, const float* __restrict__ W,
    const float* __restrict__ scale, const float* __restrict__ bias,
    float* __restrict__ Y, int M, int K, int N, int act) {
  int lane = threadIdx.x & 31;
  int wave = threadIdx.x >> 5;
  int tM = blockIdx.x;
  int tilesN = N >> 4;
  int tN0 = blockIdx.y * 16 + wave;  // this wave: tiles tN0, +4, +8, +12
  if (tN0 >= tilesN) return;         // wave-uniform exit
  bool v1 = (tN0 + 4) < tilesN;
  bool v2 = (tN0 + 8) < tilesN;
  bool v3 = (tN0 + 12) < tilesN;
  int t1 = v1 ? tN0 + 4 : tN0;   // clamp: redundant WMMAs instead of branches
  int t2 = v2 ? tN0 + 8 : tN0;
  int t3 = v3 ? tN0 + 12 : tN0;
  int mRow = tM * 16 + (lane & 15);
  int kHi = (lane >> 4) << 1;
  int nl = lane & 15;
  const float* Arow = A + (size_t)mRow * K;
  const float* B0 = W + (size_t)(tN0 * 16 + nl);
  const float* B1 = W + (size_t)(t1 * 16 + nl);
  const float* B2 = W + (size_t)(t2 * 16 + nl);
  const float* B3 = W + (size_t)(t3 * 16 + nl);
  v8f acc0 = {}, acc1 = {}, acc2 = {}, acc3 = {};
  int Kf = K & ~3;
  for (int k0 = 0; k0 < Kf; k0 += 4) {
    int ka = k0 + kHi;
    v2f a, b;
    a.x = Arow[ka];
    a.y = Arow[ka + 1];
    size_t r0 = (size_t)ka * N;
    size_t r1 = r0 + N;
    b.x = B0[r0]; b.y = B0[r1];
    acc0 = wmma4(a, b, acc0);
    b.x = B1[r0]; b.y = B1[r1];
    acc1 = wmma4(a, b, acc1);
    b.x = B2[r0]; b.y = B2[r1];
    acc2 = wmma4(a, b, acc2);
    b.x = B3[r0]; b.y = B3[r1];
    acc3 = wmma4(a, b, acc3);
  }
  if (Kf < K) {  // masked tail (K - Kf in {1,2,3}); per-lane value masks only
    int ka = Kf + kHi;
    int kc0 = ka < K ? ka : K - 1;
    int kc1 = ka + 1 < K ? ka + 1 : K - 1;
    float m0 = ka < K ? 1.f : 0.f;
    float m1 = ka + 1 < K ? 1.f : 0.f;
    v2f a, b;
    a.x = Arow[kc0] * m0;
    a.y = Arow[kc1] * m1;
    size_t r0 = (size_t)kc0 * N;
    size_t r1 = (size_t)kc1 * N;
    b.x = B0[r0] * m0; b.y = B0[r1] * m1;
    acc0 = wmma4(a, b, acc0);
    b.x = B1[r0] * m0; b.y = B1[r1] * m1;
    acc1 = wmma4(a, b, acc1);
    b.x = B2[r0] * m0; b.y = B2[r1] * m1;
    acc2 = wmma4(a, b, acc2);
    b.x = B3[r0] * m0; b.y = B3[r1] * m1;
    acc3 = wmma4(a, b, acc3);
  }
  int mBase = tM * 16 + ((lane >> 4) << 3);
  store_tile(Y, acc0, scale, bias, N, act, mBase, tN0 * 16 + nl);
  if (v1) store_tile(Y, acc1, scale, bias, N, act, mBase, t1 * 16 + nl);
  if (v2) store_tile(Y, acc2, scale, bias, N, act, mBase, t2 * 16 + nl);
  if (v3) store_tile(Y, acc3, scale, bias, N, act, mBase, t3 * 16 + nl);
}

// ------------------- batched sim = sigmoid(X @ X^T) ------------------------
// X: (B, R, K) row-major; Y: (B, R, R). R % 256 == 0, K % 4 == 0.
__global__ __launch_bounds__(128) void gemm_nt_sigmoid_kernel(
    const float* __restrict__ X, float* __restrict__ Y, int R, int K) {
  int z = blockIdx.z;
  const float* Xb = X + (size_t)z * R * K;
  float* Yb = Y + (size_t)z * R * R;
  int lane = threadIdx.x & 31;
  int wave = threadIdx.x >> 5;
  int tM = blockIdx.x;
  int tN0 = blockIdx.y * 16 + wave;  // tiles tN0, +4, +8, +12 (all valid)
  int mRow = tM * 16 + (lane & 15);
  int kHi = (lane >> 4) << 1;
  int nl = lane & 15;
  const float* Arow = Xb + (size_t)mRow * K;
  const float* B0 = Xb + (size_t)(tN0 * 16 + nl) * K;
  const float* B1 = B0 + (size_t)64 * K;
  const float* B2 = B0 + (size_t)128 * K;
  const float* B3 = B0 + (size_t)192 * K;
  v8f acc0 = {}, acc1 = {}, acc2 = {}, acc3 = {};
  for (int k0 = 0; k0 < K; k0 += 4) {
    int ka = k0 + kHi;
    v2f a, b;
    a.x = Arow[ka];
    a.y = Arow[ka + 1];
    b.x = B0[ka]; b.y = B0[ka + 1];
    acc0 = wmma4(a, b, acc0);
    b.x = B1[ka]; b.y = B1[ka + 1];
    acc1 = wmma4(a, b, acc1);
    b.x = B2[ka]; b.y = B2[ka + 1];
    acc2 = wmma4(a, b, acc2);
    b.x = B3[ka]; b.y = B3[ka + 1];
    acc3 = wmma4(a, b, acc3);
  }
  int mBase = tM * 16 + ((lane >> 4) << 3);
#pragma unroll
  for (int v = 0; v < 8; ++v) {
    size_t row = (size_t)(mBase + v) * R;
    Yb[row + tN0 * 16 + nl] = 1.f / (1.f + expf(-acc0[v]));
    Yb[row + (tN0 + 4) * 16 + nl] = 1.f / (1.f + expf(-acc1[v]));
    Yb[row + (tN0 + 8) * 16 + nl] = 1.f / (1.f + expf(-acc2[v]));
    Yb[row + (tN0 + 12) * 16 + nl] = 1.f / (1.f + expf(-acc3[v]));
  }
}

// ---------------------------- BN fold --------------------------------------
__global__ void fold_bn_kernel(const float* __restrict__ b,
                               const float* __restrict__ g,
                               const float* __restrict__ be,
                               const float* __restrict__ mu,
                               const float* __restrict__ var,
                               float* __restrict__ scale,
                               float* __restrict__ bias, int C) {
  int c = blockIdx.x * blockDim.x + threadIdx.x;
  if (c >= C) return;
  float s = g[c] * rsqrtf(var[c] + 1e-5f);
  scale[c] = s;
  bias[c] = be[c] + (b[c] - mu[c]) * s;
}

// ---------------------------- FPS ------------------------------------------
__global__ void fps_kernel(const float* __restrict__ xyz, int* __restrict__ out,
                           int N, int npoint) {
  int b = blockIdx.x;
  const float* P = xyz + (size_t)b * N * 3;
  __shared__ float dist[512];
  __shared__ float rv[256];
  __shared__ int ri[256];
  __shared__ int cur;
  int t = threadIdx.x;
  for (int j = t; j < N; j += 256) dist[j] = 1e10f;
  if (t == 0) cur = 0;
  __syncthreads();
  for (int i = 0; i < npoint; ++i) {
    int c = cur;
    if (t == 0) out[b * npoint + i] = c;
    float cx = P[c * 3], cy = P[c * 3 + 1], cz = P[c * 3 + 2];
    float best = -1.f;
    int bi = 0;
    for (int j = t; j < N; j += 256) {
      float dx = P[j * 3] - cx, dy = P[j * 3 + 1] - cy, dz = P[j * 3 + 2] - cz;
      float d = dx * dx + dy * dy + dz * dz;
      float nd = fminf(dist[j], d);
      dist[j] = nd;
      if (nd > best) { best = nd; bi = j; }
    }
    rv[t] = best;
    ri[t] = bi;
    __syncthreads();
    for (int s = 128; s > 0; s >>= 1) {
      if (t < s) {
        if (rv[t + s] > rv[t] || (rv[t + s] == rv[t] && ri[t + s] < ri[t])) {
          rv[t] = rv[t + s];
          ri[t] = ri[t + s];
        }
      }
      __syncthreads();
    }
    if (t == 0) cur = ri[0];
    __syncthreads();
  }
}

// ---------------------------- gather centroids -----------------------------
__global__ void gather_xyz_kernel(const float* __restrict__ xyz,
                                  const int* __restrict__ idx,
                                  float* __restrict__ out, int N, int S,
                                  int total) {
  int t = blockIdx.x * blockDim.x + threadIdx.x;
  if (t >= total) return;
  int b = t / S;
  int j = idx[t];
  const float* p = xyz + ((size_t)b * N + j) * 3;
  float* o = out + (size_t)t * 3;
  o[0] = p[0]; o[1] = p[1]; o[2] = p[2];
}

// ---------------------------- ball query -----------------------------------
__global__ void ball_query_kernel(const float* __restrict__ xyz,
                                  const float* __restrict__ new_xyz,
                                  int* __restrict__ idx, int N, int S, float r2,
                                  int K, int total) {
  int t = blockIdx.x * blockDim.x + threadIdx.x;
  if (t >= total) return;
  int b = t / S;
  const float* q = new_xyz + (size_t)t * 3;
  const float* P = xyz + (size_t)b * N * 3;
  int* o = idx + (size_t)t * K;
  int cnt = 0, first = 0;
  bool found = false;
  for (int j = 0; j < N && cnt < K; ++j) {
    float dx = P[j * 3] - q[0], dy = P[j * 3 + 1] - q[1],
          dz = P[j * 3 + 2] - q[2];
    float d = dx * dx + dy * dy + dz * dz;
    if (d <= r2) {
      if (!found) { found = true; first = j; }
      o[cnt++] = j;
    }
  }
  for (; cnt < K; ++cnt) o[cnt] = first;
}

// ---------------------------- group + concat -------------------------------
__global__ void group_concat_kernel(const float* __restrict__ xyz,
                                    const float* __restrict__ new_xyz,
                                    const float* __restrict__ points,
                                    const int* __restrict__ idx,
                                    float* __restrict__ g, int N, int S, int K,
                                    int Cp, int total) {
  int t = blockIdx.x * blockDim.x + threadIdx.x;
  if (t >= total) return;
  int bs = t / K;
  int s = bs % S;
  int b = bs / S;
  int j = idx[t];
  int Ct = Cp + 3;
  float* o = g + (size_t)t * Ct;
  const float* ps = points + ((size_t)b * N + j) * Cp;
  for (int c = 0; c < Cp; ++c) o[c] = ps[c];
  const float* pj = xyz + ((size_t)b * N + j) * 3;
  const float* q = new_xyz + ((size_t)b * S + s) * 3;
  o[Cp + 0] = pj[0] - q[0];
  o[Cp + 1] = pj[1] - q[1];
  o[Cp + 2] = pj[2] - q[2];
}

// ---------------------------- channel copy / broadcast ---------------------
__global__ void copy_chan_kernel(const float* __restrict__ src,
                                 float* __restrict__ dst, int csrc, int cdst,
                                 int coff, int total) {
  int t = blockIdx.x * blockDim.x + threadIdx.x;
  if (t >= total) return;
  int r = t / csrc, c = t % csrc;
  dst[(size_t)r * cdst + coff + c] = src[t];
}

__global__ void bcast_chan_kernel(const float* __restrict__ src,
                                  float* __restrict__ dst, int rowsPer, int C,
                                  int cdst, int coff, int total) {
  int t = blockIdx.x * blockDim.x + threadIdx.x;
  if (t >= total) return;
  int c = t % C;
  int r = t / C;
  int b = r / rowsPer;
  dst[(size_t)r * cdst + coff + c] = src[(size_t)b * C + c];
}

__global__ void pe_fill_kernel(const int* __restrict__ parts,
                               float* __restrict__ dst, int rowsPer, int ctot,
                               int coff, int total) {
  int t = blockIdx.x * blockDim.x + threadIdx.x;
  if (t >= total) return;
  int b = t / rowsPer;
  dst[(size_t)t * ctot + coff] = (float)parts[b];
}

// ---------------------------- max-pool over K ------------------------------
__global__ void maxpool_kernel(const float* __restrict__ g,
                               float* __restrict__ out, int K, int C, int coff,
                               int ctot, int total) {
  int t = blockIdx.x * blockDim.x + threadIdx.x;
  if (t >= total) return;
  int c = t % C;
  int r = t / C;
  const float* base = g + (size_t)r * K * C + c;
  float m = base[0];
  for (int k = 1; k < K; ++k) m = fmaxf(m, base[(size_t)k * C]);
  out[(size_t)r * ctot + coff + c] = m;
}

// ---------------------------- 3-NN interpolation ---------------------------
__global__ void fp_interp_kernel(const float* __restrict__ xyz1,
                                 const float* __restrict__ xyz2,
                                 const float* __restrict__ pts2,
                                 float* __restrict__ dst, int N1, int S2, int C,
                                 int coff, int ctot, int total) {
  int t = blockIdx.x * blockDim.x + threadIdx.x;
  if (t >= total) return;
  int b = t / N1;
  const float* q = xyz1 + (size_t)t * 3;
  const float* P = xyz2 + (size_t)b * S2 * 3;
  float d0 = 3.4e38f, d1 = 3.4e38f, d2 = 3.4e38f;
  int i0 = 0, i1 = 0, i2 = 0;
  for (int j = 0; j < S2; ++j) {
    float dx = P[j * 3] - q[0], dy = P[j * 3 + 1] - q[1],
          dz = P[j * 3 + 2] - q[2];
    float d = dx * dx + dy * dy + dz * dz;
    if (d < d0) { d2 = d1; i2 = i1; d1 = d0; i1 = i0; d0 = d; i0 = j; }
    else if (d < d1) { d2 = d1; i2 = i1; d1 = d; i1 = j; }
    else if (d < d2) { d2 = d; i2 = j; }
  }
  float w0 = 1.f / (d0 + 1e-8f), w1 = 1.f / (d1 + 1e-8f),
        w2 = 1.f / (d2 + 1e-8f);
  float inv = 1.f / (w0 + w1 + w2);
  w0 *= inv; w1 *= inv; w2 *= inv;
  const float* p0 = pts2 + ((size_t)b * S2 + i0) * C;
  const float* p1 = pts2 + ((size_t)b * S2 + i1) * C;
  const float* p2 = pts2 + ((size_t)b * S2 + i2) * C;
  float* o = dst + (size_t)t * ctot + coff;
  for (int c = 0; c < C; ++c) o[c] = w0 * p0[c] + w1 * p1[c] + w2 * p2[c];
}

// ---------------------------- masked softmax -------------------------------
__global__ void masked_softmax_kernel(const float* __restrict__ y,
                                      const int* __restrict__ parts,
                                      float* __restrict__ z, int N, int C,
                                      int total) {
  int t = blockIdx.x * blockDim.x + threadIdx.x;
  if (t >= total) return;
  int b = t / N;
  int p = parts[b];
  const float* yr = y + (size_t)t * C;
  float* zr = z + (size_t)t * C;
  float sum = 0.f;
  for (int c = 0; c < C; ++c) {
    float v = (c < p) ? expf(yr[c]) : 0.f;
    zr[c] = v;
    sum += v;
  }
  float inv = 1.f / (sum + 1e-5f);
  for (int c = 0; c < C; ++c) zr[c] *= inv;
}

// ===========================================================================
extern "C" void kernel_launch(void* const* d_in, const int* in_sizes, int n_in,
                              void* d_out, int out_size, void* d_ws,
                              size_t ws_size, hipStream_t stream) {
  (void)in_sizes; (void)n_in; (void)out_size; (void)ws_size;
  const int B = 16, N0 = 512;
  auto F = [&](int i) { return (const float*)d_in[i]; };
  const float* xyz = F(0);
  const int* parts = (const int*)d_in[1];

  // params flattened in pytree order (sorted keys), 6 arrays/layer
  // (W, b, g, be, mu, var):
  const int PB_FP1 = 2;    // 3 layers
  const int PB_FP2 = 20;   // 2 layers
  const int PB_FP3 = 32;   // 2 layers
  const int PB_RED = 44;   // 3 layers
  const int PB_REDL = 62;  // (W, b)
  const int PB_SA1 = 64;   // 3 branches x 3 layers
  const int PB_SA2 = 118;  // 2 branches x 3 layers
  const int PB_SA3 = 154;  // 3 layers

  // ---- workspace bump allocator ----
  char* wsb = (char*)d_ws;
  size_t off = 0;
  auto alloc = [&](size_t bytes) -> void* {
    void* p = wsb + off;
    off += (bytes + 255) & ~(size_t)255;
    return p;
  };
  int* l1_idx = (int*)alloc((size_t)B * 256 * sizeof(int));
  float* l1_xyz = (float*)alloc((size_t)B * 256 * 3 * sizeof(float));
  int* l2_idx = (int*)alloc((size_t)B * 64 * sizeof(int));
  float* l2_xyz = (float*)alloc((size_t)B * 64 * 3 * sizeof(float));
  int* ballIdx = (int*)alloc((size_t)B * 256 * 128 * sizeof(int));
  float* l1_pts = (float*)alloc((size_t)B * 256 * 320 * sizeof(float));
  float* l2_pts = (float*)alloc((size_t)B * 64 * 512 * sizeof(float));
  float* l3_pts = (float*)alloc((size_t)B * 1024 * sizeof(float));
  float* l2p_new = (float*)alloc((size_t)B * 64 * 256 * sizeof(float));
  float* l1p_new = (float*)alloc((size_t)B * 256 * 128 * sizeof(float));
  float* l0_pts = (float*)alloc((size_t)B * 512 * 128 * sizeof(float));
  float* scaleTmp = (float*)alloc(1024 * sizeof(float));
  float* biasTmp = (float*)alloc(1024 * sizeof(float));
  float* bufA = (float*)alloc((size_t)67108864 * sizeof(float));
  float* bufB = (float*)alloc((size_t)67108864 * sizeof(float));

  auto gemm = [&](const float* Ain, const float* W, const float* sc,
                  const float* bi, float* Y, int M, int K, int N, int act) {
    int tilesN = (N + 15) / 16;
    dim3 grid(M / 16, (tilesN + 15) / 16, 1);
    gemm_affine_act_kernel<<<grid, 128, 0, stream>>>(Ain, W, sc, bi, Y, M, K,
                                                     N, act);
  };

  auto mlp = [&](const float* in, int M, int cin, const int* ch, int nl,
                 int pbase, float* finalOut) -> const float* {
    const float* cur = in;
    int c = cin;
    for (int l = 0; l < nl; ++l) {
      int cout = ch[l];
      fold_bn_kernel<<<(cout + 255) / 256, 256, 0, stream>>>(
          F(pbase + l * 6 + 1), F(pbase + l * 6 + 2), F(pbase + l * 6 + 3),
          F(pbase + l * 6 + 4), F(pbase + l * 6 + 5), scaleTmp, biasTmp, cout);
      float* out;
      if (l == nl - 1 && finalOut) out = finalOut;
      else out = (cur == bufA) ? bufB : bufA;
      gemm(cur, F(pbase + l * 6 + 0), scaleTmp, biasTmp, out, M, c, cout, 1);
      cur = out;
      c = cout;
    }
    return cur;
  };

  // ======================= SA1 (MSG, npoint=256) ==========================
  fps_kernel<<<B, 256, 0, stream>>>(xyz, l1_idx, N0, 256);
  gather_xyz_kernel<<<(B * 256 + 127) / 128, 128, 0, stream>>>(
      xyz, l1_idx, l1_xyz, N0, 256, B * 256);
  {
    const float radii[3] = {0.1f, 0.2f, 0.4f};
    const int Ks[3] = {32, 64, 128};
    const int ch0[3] = {32, 32, 64}, ch1[3] = {64, 64, 128},
              ch2[3] = {64, 96, 128};
    const int* chs[3] = {ch0, ch1, ch2};
    const int coffs[3] = {0, 64, 192};
    for (int br = 0; br < 3; ++br) {
      int K = Ks[br];
      int totQ = B * 256;
      ball_query_kernel<<<(totQ + 127) / 128, 128, 0, stream>>>(
          xyz, l1_xyz, ballIdx, N0, 256, radii[br] * radii[br], K, totQ);
      int totG = B * 256 * K;
      group_concat_kernel<<<(totG + 127) / 128, 128, 0, stream>>>(
          xyz, l1_xyz, xyz, ballIdx, bufA, N0, 256, K, 3, totG);
      const float* last =
          mlp(bufA, totG, 6, chs[br], 3, PB_SA1 + br * 18, nullptr);
      int cout = chs[br][2];
      int totP = B * 256 * cout;
      maxpool_kernel<<<(totP + 127) / 128, 128, 0, stream>>>(
          last, l1_pts, K, cout, coffs[br], 320, totP);
    }
  }

  // ======================= SA2 (MSG, npoint=64) ===========================
  fps_kernel<<<B, 256, 0, stream>>>(l1_xyz, l2_idx, 256, 64);
  gather_xyz_kernel<<<(B * 64 + 127) / 128, 128, 0, stream>>>(
      l1_xyz, l2_idx, l2_xyz, 256, 64, B * 64);
  {
    const float radii[2] = {0.4f, 0.8f};
    const int Ks[2] = {64, 128};
    const int ch0[3] = {128, 128, 256}, ch1[3] = {128, 192, 256};
    const int* chs[2] = {ch0, ch1};
    const int coffs[2] = {0, 256};
    for (int br = 0; br < 2; ++br) {
      int K = Ks[br];
      int totQ = B * 64;
      ball_query_kernel<<<(totQ + 127) / 128, 128, 0, stream>>>(
          l1_xyz, l2_xyz, ballIdx, 256, 64, radii[br] * radii[br], K, totQ);
      int totG = B * 64 * K;
      group_concat_kernel<<<(totG + 127) / 128, 128, 0, stream>>>(
          l1_xyz, l2_xyz, l1_pts, ballIdx, bufA, 256, 64, K, 320, totG);
      const float* last =
          mlp(bufA, totG, 323, chs[br], 3, PB_SA2 + br * 18, nullptr);
      int totP = B * 64 * 256;
      maxpool_kernel<<<(totP + 127) / 128, 128, 0, stream>>>(
          last, l2_pts, K, 256, coffs[br], 512, totP);
    }
  }

  // ======================= SA3 (group all) ================================
  {
    int rows = B * 64;
    copy_chan_kernel<<<(rows * 3 + 127) / 128, 128, 0, stream>>>(
        l2_xyz, bufA, 3, 515, 0, rows * 3);
    copy_chan_kernel<<<(rows * 512 + 127) / 128, 128, 0, stream>>>(
        l2_pts, bufA, 512, 515, 3, rows * 512);
    const int ch[3] = {256, 512, 1024};
    const float* last = mlp(bufA, rows, 515, ch, 3, PB_SA3, nullptr);
    int totP = B * 1024;
    maxpool_kernel<<<(totP + 127) / 128, 128, 0, stream>>>(
        last, l3_pts, 64, 1024, 0, 1024, totP);
  }

  // ======================= FP3 (S==1 broadcast) ===========================
  {
    int rows = B * 64;
    copy_chan_kernel<<<(rows * 512 + 127) / 128, 128, 0, stream>>>(
        l2_pts, bufA, 512, 1536, 0, rows * 512);
    bcast_chan_kernel<<<(rows * 1024 + 127) / 128, 128, 0, stream>>>(
        l3_pts, bufA, 64, 1024, 1536, 512, rows * 1024);
    const int ch[2] = {256, 256};
    mlp(bufA, rows, 1536, ch, 2, PB_FP3, l2p_new);
  }

  // ======================= FP2 ============================================
  {
    int rows = B * 256;
    copy_chan_kernel<<<(rows * 320 + 127) / 128, 128, 0, stream>>>(
        l1_pts, bufA, 320, 576, 0, rows * 320);
    fp_interp_kernel<<<(rows + 127) / 128, 128, 0, stream>>>(
        l1_xyz, l2_xyz, l2p_new, bufA, 256, 64, 256, 320, 576, rows);
    const int ch[2] = {256, 128};
    mlp(bufA, rows, 576, ch, 2, PB_FP2, l1p_new);
  }

  // ======================= FP1 ============================================
  {
    int rows = B * 512;
    copy_chan_kernel<<<(rows * 3 + 127) / 128, 128, 0, stream>>>(
        xyz, bufA, 3, 132, 0, rows * 3);
    pe_fill_kernel<<<(rows + 127) / 128, 128, 0, stream>>>(parts, bufA, 512,
                                                           132, 3, rows);
    fp_interp_kernel<<<(rows + 127) / 128, 128, 0, stream>>>(
        xyz, l1_xyz, l1p_new, bufA, 512, 256, 128, 4, 132, rows);
    const int ch[3] = {128, 128, 128};
    mlp(bufA, rows, 132, ch, 3, PB_FP1, l0_pts);
  }

  // ======================= sim = sigmoid(X @ X^T) =========================
  float* simOut = (float*)d_out;  // (B, 512, 512), symmetric -> transpose noop
  gemm_nt_sigmoid_kernel<<<dim3(512 / 16, (512 / 16) / 16, B), 128, 0,
                           stream>>>(l0_pts, simOut, 512, 128);

  // ======================= red MLP + red_last + softmax ===================
  {
    int rows = B * 512;
    const int ch[3] = {512, 256, 128};
    const float* cur = mlp(simOut, rows, 512, ch, 3, PB_RED, nullptr);
    float* ylast = (cur == bufA) ? bufB : bufA;
    gemm(cur, F(PB_REDL), nullptr, F(PB_REDL + 1), ylast, rows, 128, 64, 0);
    masked_softmax_kernel<<<(rows + 127) / 128, 128, 0, stream>>>(
        ylast, parts, (float*)d_out + (size_t)B * 512 * 512, 512, 64, rows);
  }
}